// Sparsemax_79542794321975
// MI455X (gfx1250) — compile-verified
//
#include <hip/hip_runtime.h>
#include <stdint.h>

// Sparsemax over rows of z[4096][32000], faithful to the reference module
// (ascending-sort formulation). Sort eliminated by monotonicity:
//   f(j) = 1 + j*s[j] - cumsum(s)[j]  is non-decreasing in j (ascending s),
// so k_z = D-1 iff 1 + (D-1)*max - sum > 0, else k_z = 0 (all w zero).
//   k_z = D-1: tau = (sum + 1)/(D-1)
//   k_z = 0  : tau = (min + 1)/0     (IEEE inf, matching the float division)
// out = max(z - tau, 0).
//
// CDNA5 strategy: one workgroup per row; stage the 125 KB row into LDS once
// via async global->LDS DMA (ASYNCcnt), reduce in LDS, write output with
// non-temporal B128 stores. 1 GB total HBM traffic -> ~43 us @ 23.3 TB/s.

typedef float v4f __attribute__((ext_vector_type(4)));
typedef int   v4i __attribute__((ext_vector_type(4)));

#define SPX_D    32000
#define SPX_CH   (SPX_D / 4)     // 8000 x 16B chunks per row
#define SPX_BLK  512             // 16 wave32 per block
#define SPX_NW   (SPX_BLK / 32)

#define AS_GLOBAL __attribute__((address_space(1)))
#define AS_LOCAL  __attribute__((address_space(3)))

__device__ __forceinline__ void spx_async_ld16(const void* g, void* l) {
#if __has_builtin(__builtin_amdgcn_global_load_async_to_lds_b128)
  __builtin_amdgcn_global_load_async_to_lds_b128(
      (AS_GLOBAL v4i*)g, (AS_LOCAL v4i*)l, /*offset=*/0, /*cpol=*/0);
#else
  // VDST = LDS byte address, VADDR = 64-bit global address (GV mode).
  asm volatile("global_load_async_to_lds_b128 %0, %1, off"
               :: "v"((unsigned)(uintptr_t)l), "v"(g)
               : "memory");
#endif
}

__device__ __forceinline__ void spx_wait_async0() {
#if __has_builtin(__builtin_amdgcn_s_wait_asynccnt)
  __builtin_amdgcn_s_wait_asynccnt(0);
#else
  asm volatile("s_wait_asynccnt 0" ::: "memory");
#endif
}

extern "C" __global__ __launch_bounds__(SPX_BLK)
void Sparsemax_79542794321975_kernel(const float* __restrict__ z,
                                     float* __restrict__ out,
                                     int nrows) {
  extern __shared__ unsigned char spx_smem[];
  v4f*   srow = (v4f*)spx_smem;                            // 8000 * 16 B row
  float* red  = (float*)(spx_smem + SPX_CH * sizeof(v4f)); // 3*NW + 1 floats

  const int row = blockIdx.x;
  if (row >= nrows) return;
  const int tid = threadIdx.x;

  const float* zrow = z + (size_t)row * SPX_D;

  // ---- Stage row into LDS with async DMA (CDNA5 path, ASYNCcnt) ----------
  for (int c = tid; c < SPX_CH; c += SPX_BLK)
    spx_async_ld16(zrow + 4 * c, srow + c);
  spx_wait_async0();     // each wave waits for its own async transfers
  __syncthreads();       // then makes LDS visible across waves

  // ---- Per-thread partial sum / max / min over LDS (ds_load_b128) --------
  float s  = 0.0f;
  float mx = -INFINITY;
  float mn =  INFINITY;
  for (int c = tid; c < SPX_CH; c += SPX_BLK) {
    v4f v = srow[c];
    s += (v.x + v.y) + (v.z + v.w);
    mx = fmaxf(fmaxf(mx, v.x), fmaxf(v.y, fmaxf(v.z, v.w)));
    mn = fminf(fminf(mn, v.x), fminf(v.y, fminf(v.z, v.w)));
  }

  // ---- wave32 reduction ---------------------------------------------------
  for (int off = 16; off > 0; off >>= 1) {
    s  += __shfl_xor(s, off, 32);
    mx  = fmaxf(mx, __shfl_xor(mx, off, 32));
    mn  = fminf(mn, __shfl_xor(mn, off, 32));
  }
  const int wave = tid >> 5;
  if ((tid & 31) == 0) {
    red[wave]              = s;
    red[SPX_NW + wave]     = mx;
    red[2 * SPX_NW + wave] = mn;
  }
  __syncthreads();

  // ---- Final cross-wave reduction + tau -----------------------------------
  if (tid == 0) {
    float S = 0.0f, M = -INFINITY, m = INFINITY;
    for (int w = 0; w < SPX_NW; ++w) {
      S += red[w];
      M  = fmaxf(M, red[SPX_NW + w]);
      m  = fminf(m, red[2 * SPX_NW + w]);
    }
    const float f_top = 1.0f + (float)(SPX_D - 1) * M - S;   // f(D-1)
    float m_z, k_z;
    if (f_top > 0.0f) { m_z = S; k_z = (float)(SPX_D - 1); }
    else              { m_z = m; k_z = 0.0f; }               // all w[j] == 0
    red[3 * SPX_NW] = (m_z + 1.0f) / k_z;                    // tau (IEEE /0 ok)
  }
  __syncthreads();
  const float tau = red[3 * SPX_NW];

  // ---- Emit max(z - tau, 0) with non-temporal B128 stores -----------------
  v4f* orow = (v4f*)(out + (size_t)row * SPX_D);
  for (int c = tid; c < SPX_CH; c += SPX_BLK) {
    v4f v = srow[c];
    v4f r;
    r.x = fmaxf(v.x - tau, 0.0f);
    r.y = fmaxf(v.y - tau, 0.0f);
    r.z = fmaxf(v.z - tau, 0.0f);
    r.w = fmaxf(v.w - tau, 0.0f);
    __builtin_nontemporal_store(r, orow + c);
  }
}

extern "C" void kernel_launch(void* const* d_in, const int* in_sizes, int n_in,
                              void* d_out, int out_size, void* d_ws, size_t ws_size,
                              hipStream_t stream) {
  (void)n_in; (void)d_ws; (void)ws_size; (void)out_size;
  const float* z  = (const float*)d_in[0];
  float*       po = (float*)d_out;
  const int nrows = in_sizes[0] / SPX_D;   // 4096

  const size_t shmem = SPX_CH * sizeof(v4f) + (3 * SPX_NW + 1) * sizeof(float);
  // 125 KB dynamic LDS per block (well under the 320 KB/WGP limit).
  (void)hipFuncSetAttribute((const void*)Sparsemax_79542794321975_kernel,
                            hipFuncAttributeMaxDynamicSharedMemorySize, (int)shmem);

  Sparsemax_79542794321975_kernel<<<nrows, SPX_BLK, shmem, stream>>>(z, po, nrows);
}